// MultiTexture2D_1047972021061
// MI455X (gfx1250) — compile-verified
//
#include <hip/hip_runtime.h>
#include <stdint.h>
#include <stddef.h>

// MultiTexture2D: per-pixel bilinear texture sample with wrap, material select.
// uv:       [1,2048,2048,2] f32
// f_mat:    [1,2048,2048]   int (harness delivers integer inputs as int32)
// textures: [4,1024,1024,3] f32  (48 MB total -> resident in 192 MB L2)
// out:      [1,2048,2048,3] f32

#define TEX_W 1024
#define TEX_H 1024
#define TEX_C 3

typedef float v2f __attribute__((ext_vector_type(2)));
typedef float v3f __attribute__((ext_vector_type(3)));

__global__ __launch_bounds__(256) void multitex_bilinear_kernel(
    const float* __restrict__ uv,
    const int*   __restrict__ f_mat,
    const float* __restrict__ tex,
    float*       __restrict__ out,
    int npix)
{
    int i = blockIdx.x * blockDim.x + threadIdx.x;
    if (i >= npix) return;

    // --- Streaming inputs: non-temporal (do not pollute L2 texture residency)
    v2f uvv = __builtin_nontemporal_load((const v2f*)uv + i);   // global_load_b64 th:NT
    int mat = __builtin_nontemporal_load(f_mat + i);            // global_load_b32 th:NT

    // --- Bilinear footprint with wrap (W,H are powers of two)
    float x = uvv.x * (float)TEX_W - 0.5f;
    float y = uvv.y * (float)TEX_H - 0.5f;
    float xf = floorf(x);
    float yf = floorf(y);
    float fx = x - xf;
    float fy = y - yf;
    int x0 = ((int)xf + TEX_W) & (TEX_W - 1);
    int y0 = ((int)yf + TEX_H) & (TEX_H - 1);
    int x1 = (x0 + 1) & (TEX_W - 1);
    int y1 = (y0 + 1) & (TEX_H - 1);

    // Select-then-sample: only the chosen material's texture is read (4x less
    // gather traffic than the reference's sample-all-then-select).
    const float* base = tex + (size_t)mat * (size_t)(TEX_W * TEX_H * TEX_C);
    uint64_t a00 = (uint64_t)(uintptr_t)(base + (size_t)(((y0 << 10) + x0) * 3));
    uint64_t a01 = (uint64_t)(uintptr_t)(base + (size_t)(((y0 << 10) + x1) * 3));
    uint64_t a10 = (uint64_t)(uintptr_t)(base + (size_t)(((y1 << 10) + x0) * 3));
    uint64_t a11 = (uint64_t)(uintptr_t)(base + (size_t)(((y1 << 10) + x1) * 3));

    // --- Four 96-bit corner gathers with HT (high-priority temporal) hint:
    // keeps the 48 MB texture set pinned in the 192 MB L2 while NT streams
    // flow around it. Loads + split-counter wait fused in one asm block so
    // the destination VGPRs are never read before LOADcnt retires (ISA 7.1).
    v3f t00, t01, t10, t11;
    asm volatile(
        "global_load_b96 %0, %4, off th:TH_LOAD_HT\n\t"
        "global_load_b96 %1, %5, off th:TH_LOAD_HT\n\t"
        "global_load_b96 %2, %6, off th:TH_LOAD_HT\n\t"
        "global_load_b96 %3, %7, off th:TH_LOAD_HT\n\t"
        "s_wait_loadcnt 0x0"
        : "=&v"(t00), "=&v"(t01), "=&v"(t10), "=&v"(t11)
        : "v"(a00), "v"(a01), "v"(a10), "v"(a11));

    // --- Blend (matches reference arithmetic: w00+w01+w10+w11 == 1 exactly
    // in the same factored form)
    float w11 = fx * fy;
    float w01 = fx * (1.0f - fy);
    float w10 = (1.0f - fx) * fy;
    float w00 = (1.0f - fx) * (1.0f - fy);

    float r = t00.x * w00 + t01.x * w01 + t10.x * w10 + t11.x * w11;
    float g = t00.y * w00 + t01.y * w01 + t10.y * w10 + t11.y * w11;
    float b = t00.z * w00 + t01.z * w01 + t10.z * w10 + t11.z * w11;

    // --- Non-temporal streaming store of the result (one-touch data)
    size_t o = (size_t)i * 3u;
    __builtin_nontemporal_store(r, out + o + 0);
    __builtin_nontemporal_store(g, out + o + 1);
    __builtin_nontemporal_store(b, out + o + 2);
}

extern "C" void kernel_launch(void* const* d_in, const int* in_sizes, int n_in,
                              void* d_out, int out_size, void* d_ws, size_t ws_size,
                              hipStream_t stream) {
    (void)n_in; (void)d_ws; (void)ws_size; (void)out_size;
    const float* uv    = (const float*)d_in[0];
    const int*   f_mat = (const int*)d_in[1];
    const float* tex   = (const float*)d_in[2];
    float*       out   = (float*)d_out;

    const int npix = in_sizes[1];           // B*H*W = 4,194,304
    const int block = 256;                  // 8 wave32 per block
    const int grid = (npix + block - 1) / block;
    multitex_bilinear_kernel<<<grid, block, 0, stream>>>(uv, f_mat, tex, out, npix);
}